// ManualChebConv_26388279067324
// MI455X (gfx1250) — compile-verified
//
#include <hip/hip_runtime.h>

typedef float v2f __attribute__((ext_vector_type(2)));
typedef float v8f __attribute__((ext_vector_type(8)));

#define N_NODES 512
#define F_IN    16
#define F_OUT   8
#define K_CHEB  5
#define TPW     4            // row-tiles per wave (32 tiles / 8 waves)
#define BPW     2            // batches per workgroup (A-fragment reuse)
#define BLOCK   256
#define ZSTRIDE 8192         // floats per batch Z buffer [512][16]

// LDS layout (floats)
#define ZA_OFF  0                       // [BPW][8192]
#define ZB_OFF  (BPW * ZSTRIDE)         // [BPW][8192]
#define WP_OFF  (2 * BPW * ZSTRIDE)     // padded W: [5][16][16]
#define LDS_FLOATS (2 * BPW * ZSTRIDE + K_CHEB * 256)

__device__ __forceinline__ v8f wmma_f32(v2f a, v2f b, v8f c) {
    // D = A(16x4 f32) * B(4x16 f32) + C(16x16 f32)
    return __builtin_amdgcn_wmma_f32_16x16x4_f32(
        /*neg_a=*/false, a, /*neg_b=*/false, b,
        /*c_mod=*/(short)0, c, /*reuse_a=*/false, /*reuse_b=*/false);
}

__global__ void __launch_bounds__(BLOCK)
cheb_conv_kernel(const float* __restrict__ x,
                 const float* __restrict__ L,
                 const float* __restrict__ W,
                 const float* __restrict__ bias,
                 float* __restrict__ out)
{
    __shared__ __align__(16) float lds[LDS_FLOATS];
    float* zA = lds + ZA_OFF;
    float* zB = lds + ZB_OFF;
    float* wp = lds + WP_OFF;

    const int tid   = threadIdx.x;
    const int wave  = tid >> 5;
    const int lane  = tid & 31;
    const int laneM = lane & 15;   // row (A) / col (B,C,D) within tile
    const int laneH = lane >> 4;   // K-half selector
    const int bidx  = blockIdx.x;  // handles batches [bidx*BPW, bidx*BPW+1]

    // ------- stage Z0 = x[bidx*BPW .. +1] into zA (coalesced b128) ---------
    {
        const float4* xs  = (const float4*)(x + (size_t)bidx * (BPW * N_NODES * F_IN));
        float4*       za4 = (float4*)zA;
        #pragma unroll
        for (int i = 0; i < BPW * 8; ++i)
            za4[tid + i * BLOCK] = xs[tid + i * BLOCK];
    }
    // ------- stage W padded to [5][16][16] (cols 8..15 = 0) ----------------
    for (int i = tid; i < K_CHEB * 256; i += BLOCK) {
        int k  = i >> 8;
        int r  = i & 255;
        int fi = r >> 4;
        int fo = r & 15;
        wp[i] = (fo < F_OUT) ? W[(k * F_IN + fi) * F_OUT + fo] : 0.0f;
    }
    __syncthreads();

    // output accumulators: [batch][tile] 16x16 f32 D — only cols 0..7 used
    v8f oacc[BPW][TPW];
    #pragma unroll
    for (int bp = 0; bp < BPW; ++bp)
        #pragma unroll
        for (int t = 0; t < TPW; ++t)
            oacc[bp][t] = (v8f){0.f, 0.f, 0.f, 0.f, 0.f, 0.f, 0.f, 0.f};

    // oacc[bp] += Zk_tile(16x16) @ Wpad[k](16x16); A from LDS zc
    auto project = [&](const float* zc, int k) {
        const float* wk = wp + k * 256;
        #pragma unroll
        for (int kk = 0; kk < 4; ++kk) {
            int brow = kk * 4 + 2 * laneH;
            v2f bfr;
            bfr.x = wk[brow * 16 + laneM];
            bfr.y = wk[(brow + 1) * 16 + laneM];
            #pragma unroll
            for (int bp = 0; bp < BPW; ++bp) {
                #pragma unroll
                for (int t = 0; t < TPW; ++t) {
                    int m = wave * TPW + t;
                    const float* ap = zc + bp * ZSTRIDE +
                                      (m * 16 + laneM) * F_IN + kk * 4 + 2 * laneH;
                    v2f a; a.x = ap[0]; a.y = ap[1];
                    oacc[bp][t] = wmma_f32(a, bfr, oacc[bp][t]);
                }
            }
        }
    };

    // zdst(own rows) = first ? L@zsrc : 2*(L@zsrc) - zprev(own rows), per batch
    auto lstep = [&](float* zdst, const float* zsrc, const float* zprev, bool first) {
        v8f acc[BPW][TPW];
        #pragma unroll
        for (int bp = 0; bp < BPW; ++bp) {
            #pragma unroll
            for (int t = 0; t < TPW; ++t) {
                if (first) {
                    acc[bp][t] = (v8f){0.f, 0.f, 0.f, 0.f, 0.f, 0.f, 0.f, 0.f};
                } else {
                    int m = wave * TPW + t;
                    #pragma unroll
                    for (int j = 0; j < 8; ++j)
                        acc[bp][t][j] = -0.5f * zprev[bp * ZSTRIDE +
                                        (m * 16 + j + 8 * laneH) * F_IN + laneM];
                }
            }
        }
        // per-lane L base: row = wave*64 + laneM (tile t adds 16 rows), col = 2*laneH
        const float* lp = L + (size_t)(wave * 64 + laneM) * N_NODES + 2 * laneH;
        #pragma unroll 2
        for (int kk = 0; kk < N_NODES / 4; ++kk) {
            int brow = kk * 4 + 2 * laneH;
            v2f bv[BPW];                       // B frags (one per batch), reused by 4 tiles
            #pragma unroll
            for (int bp = 0; bp < BPW; ++bp) {
                bv[bp].x = zsrc[bp * ZSTRIDE + brow * F_IN + laneM];
                bv[bp].y = zsrc[bp * ZSTRIDE + (brow + 1) * F_IN + laneM];
            }
            const float* lpk = lp + kk * 4;
            #pragma unroll
            for (int t = 0; t < TPW; ++t) {
                v2f a = *(const v2f*)(lpk + t * 16 * N_NODES);  // global_load_b64
                #pragma unroll
                for (int bp = 0; bp < BPW; ++bp)                // A reused BPW times
                    acc[bp][t] = wmma_f32(a, bv[bp], acc[bp][t]);
            }
        }
        float s = first ? 1.0f : 2.0f;
        #pragma unroll
        for (int bp = 0; bp < BPW; ++bp) {
            #pragma unroll
            for (int t = 0; t < TPW; ++t) {
                int m = wave * TPW + t;
                #pragma unroll
                for (int j = 0; j < 8; ++j)
                    zdst[bp * ZSTRIDE + (m * 16 + j + 8 * laneH) * F_IN + laneM] =
                        s * acc[bp][t][j];
            }
        }
    };

    // ---------------- Chebyshev recurrence ---------------------------------
    project(zA, 0);                        // out += Z0 @ W0
    lstep(zB, zA, nullptr, true);          // Z1 = L @ Z0
    __syncthreads();
    project(zB, 1);                        // out += Z1 @ W1
    lstep(zA, zB, zA, false);              // Z2 = 2 L Z1 - Z0
    __syncthreads();
    project(zA, 2);
    lstep(zB, zA, zB, false);              // Z3 = 2 L Z2 - Z1
    __syncthreads();
    project(zB, 3);
    lstep(zA, zB, zA, false);              // Z4 = 2 L Z3 - Z2
    __syncthreads();
    project(zA, 4);

    // ------- add bias, stage compact [512][8] per batch into zB ------------
    float bval = bias[laneM & 7];
    if (laneM < F_OUT) {
        #pragma unroll
        for (int bp = 0; bp < BPW; ++bp) {
            #pragma unroll
            for (int t = 0; t < TPW; ++t) {
                int m = wave * TPW + t;
                #pragma unroll
                for (int j = 0; j < 8; ++j)
                    zB[bp * ZSTRIDE + (m * 16 + j + 8 * laneH) * F_OUT + laneM] =
                        oacc[bp][t][j] + bval;
            }
        }
    }
    __syncthreads();

    // ---------------- coalesced b128 stores --------------------------------
    #pragma unroll
    for (int bp = 0; bp < BPW; ++bp) {
        float4*       os  = (float4*)(out + ((size_t)bidx * BPW + bp) * (N_NODES * F_OUT));
        const float4* zb4 = (const float4*)(zB + bp * ZSTRIDE);
        #pragma unroll
        for (int i = 0; i < 4; ++i)
            os[tid + i * BLOCK] = zb4[tid + i * BLOCK];
    }
}

extern "C" void kernel_launch(void* const* d_in, const int* in_sizes, int n_in,
                              void* d_out, int out_size, void* d_ws, size_t ws_size,
                              hipStream_t stream) {
    const float* x    = (const float*)d_in[0];
    const float* L    = (const float*)d_in[1];
    const float* W    = (const float*)d_in[2];
    const float* bias = (const float*)d_in[3];
    float* out = (float*)d_out;

    int B = in_sizes[0] / (N_NODES * F_IN);   // 1024
    cheb_conv_kernel<<<dim3(B / BPW), dim3(BLOCK), 0, stream>>>(x, L, W, bias, out);
}